// LearnedPosMap_83356725280796
// MI455X (gfx1250) — compile-verified
//
#include <hip/hip_runtime.h>
#include <hip/hip_bf16.h>
#include <stdint.h>

// Problem constants (match reference)
#define B_     32
#define W_     16
#define N_     256
#define DIM_   256
#define GAMMA_ 4
#define CH_    64      // W_*GAMMA_ : bias_table row length
#define KHALF  128     // K staged in two halves
#define LDSS   130     // LDS row stride (floats): even (8B-aligned float2) + bank spread

typedef __attribute__((ext_vector_type(2))) float v2f;
typedef __attribute__((ext_vector_type(8))) float v8f;

__global__ __launch_bounds__(256)
void LearnedPosMap_wmma_f32(const float* __restrict__ x,
                            const float* __restrict__ bias_table,
                            const float* __restrict__ token_bias,
                            const int*   __restrict__ rel_index,
                            float* __restrict__ out)
{
    // X^T slice for this (b,w,s): Xt[v][k], v in [0,64), k in [0,128)
    __shared__ float xt[64 * LDSS];

    const int tid = threadIdx.x;
    const int bid = blockIdx.x;
    const int s = bid & 3;
    const int w = (bid >> 2) & 15;
    const int b = bid >> 6;

    const int chan  = s * W_ + w;        // bias_table column: (s outer, w inner)
    const int tbrow = w * GAMMA_ + s;    // token_bias row:    (w outer, s inner)

    const float* __restrict__ xw   = x   + (size_t)(b * W_ + w) * N_ * DIM_;
    float*       __restrict__ outw = out + (size_t)(b * W_ + w) * N_ * DIM_;
    // Fold the channel into the gather base: gathers become base + 32-bit voffset
    const float* __restrict__ btc  = bias_table + chan;
    const float* __restrict__ tb   = token_bias;   // row tbrow, 32-bit offsets below

    const int wave  = tid >> 5;
    const int lane  = tid & 31;
    const int half  = lane >> 4;   // 0: lanes 0-15, 1: lanes 16-31
    const int lq    = lane & 15;
    const int mbase = wave * 32;   // each wave: M-tiles [mbase, mbase+16), [mbase+16, mbase+32)

    // Per-lane rel_index row bases (hoisted so the K loop uses immediate offsets)
    const int* __restrict__ ri0 = rel_index + (uint32_t)((mbase      + lq) * N_) + 2 * half;
    const int* __restrict__ ri1 = rel_index + (uint32_t)((mbase + 16 + lq) * N_) + 2 * half;

    v8f acc[2][4] = {};            // 2 M-tiles x 4 V-tiles of 16x16 f32

    for (int kb = 0; kb < N_; kb += KHALF) {
        // ---- stage x[b,w, kb..kb+127, 4v+s] transposed into LDS ----
        for (int e = tid; e < 64 * KHALF; e += 256) {
            const int v  = e & 63;
            const int kk = e >> 6;
            xt[v * LDSS + kk] = xw[(size_t)(kb + kk) * DIM_ + 4 * v + s];
        }
        __syncthreads();

        // ---- WMMA main loop: K-step 4 (V_WMMA_F32_16X16X4_F32) ----
        for (int k0 = 0; k0 < KHALF; k0 += 4) {
            #pragma unroll
            for (int mt = 0; mt < 2; ++mt) {
                // A fragment (16x4 f32): lane l<16 holds row m, K={k0,k0+1};
                // lane l+16 holds row m, K={k0+2,k0+3}
                const int* rp = (mt == 0) ? ri0 : ri1;
                const int2 ip = *(const int2*)(rp + kb + k0);
                v2f a;
                a.x = btc[(uint32_t)ip.x << 6];   // saddr + 32-bit voffset gather
                a.y = btc[(uint32_t)ip.y << 6];
                #pragma unroll
                for (int vt = 0; vt < 4; ++vt) {
                    // B fragment (4x16 f32): lane n holds column v0+n,
                    // VGPR0 = K=k0(+2 for high half), VGPR1 = K=k0+1(+2)
                    const float2 bq =
                        *(const float2*)(&xt[(vt * 16 + lq) * LDSS + k0 + 2 * half]);
                    v2f bf; bf.x = bq.x; bf.y = bq.y;
                    acc[mt][vt] = __builtin_amdgcn_wmma_f32_16x16x4_f32(
                        /*neg_a=*/false, a, /*neg_b=*/false, bf,
                        /*c_mod=*/(short)0, acc[mt][vt],
                        /*reuse_a=*/false, /*reuse_b=*/false);
                }
            }
        }
        __syncthreads();
    }

    // ---- epilogue: C/D layout -> global, plus token bias ----
    // VGPR r: lanes 0-15 -> M = tile+r, lanes 16-31 -> M = tile+8+r; N = lq
    #pragma unroll
    for (int mt = 0; mt < 2; ++mt) {
        #pragma unroll
        for (int r = 0; r < 8; ++r) {
            const int m = mbase + mt * 16 + half * 8 + r;
            const float bias = tb[(uint32_t)(tbrow * N_ + m)];
            float* __restrict__ orow = outw + (size_t)m * DIM_ + s;
            #pragma unroll
            for (int vt = 0; vt < 4; ++vt) {
                orow[4 * (vt * 16 + lq)] = acc[mt][vt][r] + bias;
            }
        }
    }
}

extern "C" void kernel_launch(void* const* d_in, const int* in_sizes, int n_in,
                              void* d_out, int out_size, void* d_ws, size_t ws_size,
                              hipStream_t stream) {
    (void)in_sizes; (void)n_in; (void)out_size; (void)d_ws; (void)ws_size;
    const float* x          = (const float*)d_in[0];
    const float* bias_table = (const float*)d_in[1];
    const float* token_bias = (const float*)d_in[2];
    const int*   rel_index  = (const int*)d_in[3];
    float* out = (float*)d_out;

    dim3 grid(B_ * W_ * GAMMA_);   // 2048 workgroups: one (b, w, s) GEMM each
    dim3 block(256);               // 8 waves (wave32)
    LearnedPosMap_wmma_f32<<<grid, block, 0, stream>>>(x, bias_table, token_bias,
                                                       rel_index, out);
}